// GCNClassifier_22351009809268
// MI455X (gfx1250) — compile-verified
//
#include <hip/hip_runtime.h>

typedef __attribute__((ext_vector_type(2))) float v2f;
typedef __attribute__((ext_vector_type(8))) float v8f;

#define FDIM 64

// ---- degree / normalization ------------------------------------------------
__global__ void deg_init_k(float* __restrict__ deg, int n) {
    int i = blockIdx.x * blockDim.x + threadIdx.x;
    if (i < n) deg[i] = 1.0f;  // self-loop contributes 1 to every node
}

__global__ void deg_acc_k(const int* __restrict__ dst, float* __restrict__ deg, int e) {
    int i = blockIdx.x * blockDim.x + threadIdx.x;
    if (i < e) atomicAdd(&deg[dst[i]], 1.0f);
}

__global__ void dinv_k(float* __restrict__ deg, int n) {
    int i = blockIdx.x * blockDim.x + threadIdx.x;
    if (i < n) deg[i] = rsqrtf(deg[i]);   // deg >= 1 always (self-loops)
}

// ---- weight transpose: Wt[n][k] = W[k][n]  (64xN -> Nx64, tiny) ------------
__global__ void transpose_k(const float* __restrict__ W, float* __restrict__ Wt,
                            int ncols) {
    const int K = FDIM;
    int t = blockIdx.x * blockDim.x + threadIdx.x;
    if (t < K * ncols) {
        int k = t / ncols;
        int nn = t - k * ncols;
        Wt[nn * K + k] = W[t];
    }
}

// ---- dense GEMM via exact-fp32 WMMA ----------------------------------------
// One wave computes a 16-row stripe across ALL NTN column tiles:
//   C[16 x NTN*16] = A[16 x 64] * Bt^T, A loaded once per K-step,
//   all operand loads are contiguous b64 with immediate offsets.
template<int NTN>
__global__ void gemm_wmma_k(const float* __restrict__ A, const float* __restrict__ Bt,
                            float* __restrict__ C, int mtiles,
                            const float* __restrict__ bias, int addBias) {
    const int K = FDIM;
    const int ncols = NTN * 16;
    int wave = threadIdx.x >> 5;
    int lane = threadIdx.x & 31;
    int tm = blockIdx.x * (blockDim.x >> 5) + wave;
    if (tm >= mtiles) return;                 // wave-uniform guard
    int half = lane >> 4;                     // K sub-block select (ISA A/B layout)
    int l    = lane & 15;                     // M index (A) / N index (B, C, D)

    const float* ap = A  + ((size_t)tm * 16 + l) * K + half * 2;
    const float* bp = Bt + (size_t)l * K + half * 2;

    v8f acc[NTN] = {};
#pragma unroll
    for (int k0 = 0; k0 < K; k0 += 4) {
        v2f a = *(const v2f*)(ap + k0);       // A 16x4 tile operand
#pragma unroll
        for (int t = 0; t < NTN; ++t) {
            v2f b = *(const v2f*)(bp + (size_t)t * 16 * K + k0);  // B 4x16 operand
            acc[t] = __builtin_amdgcn_wmma_f32_16x16x4_f32(
                false, a, false, b, (short)0, acc[t], false, false);
        }
    }

#pragma unroll
    for (int t = 0; t < NTN; ++t) {
        int col = t * 16 + l;
        float bv = addBias ? bias[col] : 0.0f;
#pragma unroll
        for (int j = 0; j < 8; ++j) {
            int row = tm * 16 + half * 8 + j;   // D VGPR j: M=j / M=j+8 per lane half
            C[(size_t)row * ncols + col] = acc[t][j] + bv;
        }
    }
}

// ---- aggregation: out = self-loop term, then scatter-add edges -------------
__global__ void agg_init_k(const float* __restrict__ h, const float* __restrict__ dinv,
                           float* __restrict__ out, int n) {
    int t = blockIdx.x * blockDim.x + threadIdx.x;
    int i = t >> 6;
    if (i < n) {
        float d = dinv[i];
        out[t] = h[t] * d * d;   // self-loop norm = dinv[i]^2
    }
}

__global__ void agg_edge_k(const float* __restrict__ h, const int* __restrict__ src,
                           const int* __restrict__ dst, const float* __restrict__ dinv,
                           float* __restrict__ out, int e) {
    int t = blockIdx.x * blockDim.x + threadIdx.x;
    int ed = t >> 4;            // 16 lanes per edge, float4 each
    if (ed >= e) return;
    int f4 = (t & 15) << 2;
    int s = src[ed];
    int d = dst[ed];
    float w = dinv[s] * dinv[d];
    const float4 hv = *(const float4*)(h + (size_t)s * FDIM + f4);
    float* o = out + (size_t)d * FDIM + f4;
    atomicAdd(o + 0, w * hv.x);
    atomicAdd(o + 1, w * hv.y);
    atomicAdd(o + 2, w * hv.z);
    atomicAdd(o + 3, w * hv.w);
}

__global__ void bias_relu_k(float* __restrict__ h, const float* __restrict__ b, int n) {
    int t = blockIdx.x * blockDim.x + threadIdx.x;
    int i = t >> 6;
    if (i < n) {
        float v = h[t] + b[t & 63];
        h[t] = v > 0.0f ? v : 0.0f;
    }
}

// ---- host orchestration ----------------------------------------------------
extern "C" void kernel_launch(void* const* d_in, const int* in_sizes, int n_in,
                              void* d_out, int out_size, void* d_ws, size_t ws_size,
                              hipStream_t stream) {
    const float* x  = (const float*)d_in[0];
    const int*   ei = (const int*)d_in[1];
    const float* W1 = (const float*)d_in[2];
    const float* b1 = (const float*)d_in[3];
    const float* W2 = (const float*)d_in[4];
    const float* b2 = (const float*)d_in[5];
    const float* Wl = (const float*)d_in[6];
    const float* bl = (const float*)d_in[7];

    int n = in_sizes[0] / FDIM;   // 100000
    int e = in_sizes[1] / 2;      // 1000000
    const int* src = ei;
    const int* dst = ei + e;

    float* ws   = (float*)d_ws;
    float* dinv = ws;
    size_t off  = ((size_t)n + 255) & ~(size_t)255;
    float* bufA = ws + off;
    float* bufB = bufA + (size_t)n * FDIM;
    float* Wt1  = bufB + (size_t)n * FDIM;
    float* Wt2  = Wt1 + FDIM * FDIM;
    float* Wtl  = Wt2 + FDIM * FDIM;

    int nblk_n    = (n + 255) / 256;
    int nblk_e    = (e + 255) / 256;
    int nblk_nf   = (int)(((size_t)n * FDIM + 255) / 256);
    int nblk_edge = (int)(((size_t)e * 16 + 255) / 256);

    int mtiles   = n / 16;                       // 6250 (N divisible by 16)
    int gblk     = (mtiles + 3) / 4;             // 4 waves (stripes) per block

    // weight transposes (tiny)
    transpose_k<<<(FDIM * FDIM + 255) / 256, 256, 0, stream>>>(W1, Wt1, FDIM);
    transpose_k<<<(FDIM * FDIM + 255) / 256, 256, 0, stream>>>(W2, Wt2, FDIM);
    transpose_k<<<(FDIM * 32  + 255) / 256, 256, 0, stream>>>(Wl, Wtl, 32);

    // degrees -> dinv
    deg_init_k<<<nblk_n, 256, 0, stream>>>(dinv, n);
    deg_acc_k <<<nblk_e, 256, 0, stream>>>(dst, dinv, e);
    dinv_k    <<<nblk_n, 256, 0, stream>>>(dinv, n);

    // layer 1: h1 = relu(Agg(x @ W1) + b1)
    gemm_wmma_k<4><<<gblk, 128, 0, stream>>>(x, Wt1, bufA, mtiles, nullptr, 0);
    agg_init_k <<<nblk_nf, 256, 0, stream>>>(bufA, dinv, bufB, n);
    agg_edge_k <<<nblk_edge, 256, 0, stream>>>(bufA, src, dst, dinv, bufB, e);
    bias_relu_k<<<nblk_nf, 256, 0, stream>>>(bufB, b1, n);

    // layer 2: h2 = relu(Agg(h1 @ W2) + b2)
    gemm_wmma_k<4><<<gblk, 128, 0, stream>>>(bufB, Wt2, bufA, mtiles, nullptr, 0);
    agg_init_k <<<nblk_nf, 256, 0, stream>>>(bufA, dinv, bufB, n);
    agg_edge_k <<<nblk_edge, 256, 0, stream>>>(bufA, src, dst, dinv, bufB, e);
    bias_relu_k<<<nblk_nf, 256, 0, stream>>>(bufB, b2, n);

    // classifier: out = h2 @ Wl + bl
    gemm_wmma_k<2><<<gblk, 128, 0, stream>>>(bufB, Wtl, (float*)d_out, mtiles, bl, 1);
}